// ResonantLayerV2_75892072121075
// MI455X (gfx1250) — compile-verified
//
#include <hip/hip_runtime.h>
#include <hip/hip_bf16.h>
#include <math.h>

typedef float v2f __attribute__((ext_vector_type(2)));
typedef float v8f __attribute__((ext_vector_type(8)));

#define D_MODEL 1024
#define NUM_NEURONS 2048
#define LUT_SIZE 4096
#define BATCH 64
#define NB 4                 // batch rows per block in resonant core
#define INV_TWO_PI 0.15915494309189535f

// ---------------------------------------------------------------------------
// Kernel 1: x_collapsed[b,d] = concat(x_real,x_imag)[b,:] . ic_w[d,:] + ic_b[d]
//   M=64 (4 tiles), N=1024 (64 tiles), K=2048. One wave per 16x16 tile,
//   V_WMMA_F32_16X16X4_F32 accumulation.
// ---------------------------------------------------------------------------
__global__ void __launch_bounds__(128)
collapse_gemm_kernel(const float* __restrict__ xr, const float* __restrict__ xi,
                     const float* __restrict__ icw, const float* __restrict__ icb,
                     float* __restrict__ xc) {
    const int K = 2 * D_MODEL;
    int wave = threadIdx.x >> 5;
    int lane = threadIdx.x & 31;
    int tile = blockIdx.x * 4 + wave;        // 256 tiles total
    int tm = tile & 3;                       // M tile (batch)
    int tn = tile >> 2;                      // N tile (d)
    int half = lane >> 4;
    int l16  = lane & 15;
    int arow = tm * 16 + l16;                // batch row this lane feeds A with
    int bcol = tn * 16 + l16;                // output column this lane feeds B with

    v8f acc = {};
    for (int k0 = 0; k0 < K; k0 += 4) {
        int ka = k0 + half * 2;              // lanes 0-15: k0,k0+1 ; 16-31: k0+2,k0+3
        v2f a, b;
        // A = concat(x_real, x_imag)[arow, ka..ka+1]  (ka even -> never straddles 1024)
        const float* asrc = (ka < D_MODEL) ? (xr + arow * D_MODEL + ka)
                                           : (xi + arow * D_MODEL + (ka - D_MODEL));
        a.x = asrc[0]; a.y = asrc[1];
        // B[k, n] = ic_w[n, k] (row-major [1024,2048]) -> contiguous in k
        const float* bsrc = icw + (size_t)bcol * K + ka;
        b.x = bsrc[0]; b.y = bsrc[1];
        acc = __builtin_amdgcn_wmma_f32_16x16x4_f32(false, a, false, b,
                                                    (short)0, acc, false, false);
    }
    // C/D layout: VGPR r -> M = half*8 + r, N = l16
    float bias = icb[bcol];
    int mbase = tm * 16 + half * 8;
    #pragma unroll
    for (int r = 0; r < 8; ++r)
        xc[(size_t)(mbase + r) * D_MODEL + bcol] = acc[r] + bias;
}

// ---------------------------------------------------------------------------
// Kernel 2: resonant core (dominant cost).
//   theta[b,n,d] = x_collapsed[b,d]/(1+|W[n,d]|) + Bp[n,d] + t[b]
//   LDS LUT: tab4[i] = (sin[i], cos[i], sin[i+1], cos[i+1])  -> one
//   ds_load_b128 gather per element, wrap baked into entry 4095.
//   One wave per neuron; each wave processes NB=4 batch rows so the
//   per-(n,d) streams (W, Bp, attn, rcp) are loaded once and reused 4x.
//   Lanes own 4 consecutive d -> all global traffic is b128.
// ---------------------------------------------------------------------------
__global__ void __launch_bounds__(256)
resonant_core_kernel(const float* __restrict__ xc, const float* __restrict__ t,
                     const float* __restrict__ W, const float* __restrict__ Bp,
                     const float* __restrict__ ac, const float* __restrict__ as_,
                     const float* __restrict__ sin_t, const float* __restrict__ cos_t,
                     float* __restrict__ cos_sum, float* __restrict__ sin_sum) {
    __shared__ float4 tab4[LUT_SIZE];    // 64 KB: (sin_i, cos_i, sin_i+1, cos_i+1)

    int bg = blockIdx.x >> 8;            // 16 batch groups of NB=4
    int ng = blockIdx.x & 255;           // 256 neuron groups of 8
    int b0 = bg * NB;

    for (int i = threadIdx.x; i < LUT_SIZE; i += blockDim.x) {
        int ip = (i + 1) & (LUT_SIZE - 1);
        tab4[i] = make_float4(sin_t[i], cos_t[i], sin_t[ip], cos_t[ip]);
    }
    __syncthreads();

    int wave = threadIdx.x >> 5;
    int lane = threadIdx.x & 31;
    int n = ng * 8 + wave;
    const size_t nrow = (size_t)n * D_MODEL;

    float tb[NB];
    #pragma unroll
    for (int bb = 0; bb < NB; ++bb) tb[bb] = t[b0 + bb];

    float accc[NB] = {}, accs[NB] = {};

    #pragma unroll 2
    for (int it = 0; it < D_MODEL / 128; ++it) {   // 8 chunks
        int d = it * 128 + lane * 4;               // 4 consecutive d per lane
        float4 w4  = *(const float4*)(W  + nrow + d);
        float4 bp4 = *(const float4*)(Bp + nrow + d);
        float4 ac4 = *(const float4*)(ac + nrow + d);
        float4 as4 = *(const float4*)(as_ + nrow + d);
        // reciprocal wavelengths, shared across the NB batch rows
        float rw0 = __builtin_amdgcn_rcpf(1.0f + fabsf(w4.x));
        float rw1 = __builtin_amdgcn_rcpf(1.0f + fabsf(w4.y));
        float rw2 = __builtin_amdgcn_rcpf(1.0f + fabsf(w4.z));
        float rw3 = __builtin_amdgcn_rcpf(1.0f + fabsf(w4.w));
        float rw[4]  = {rw0, rw1, rw2, rw3};
        float bp[4]  = {bp4.x, bp4.y, bp4.z, bp4.w};
        float acw[4] = {ac4.x, ac4.y, ac4.z, ac4.w};
        float asw[4] = {as4.x, as4.y, as4.z, as4.w};

        #pragma unroll
        for (int bb = 0; bb < NB; ++bb) {
            float4 x4 = *(const float4*)(xc + (size_t)(b0 + bb) * D_MODEL + d);
            float xv[4] = {x4.x, x4.y, x4.z, x4.w};
            #pragma unroll
            for (int j = 0; j < 4; ++j) {
                float th   = fmaf(xv[j], rw[j], bp[j] + tb[bb]);
                float u    = th * INV_TWO_PI;
                u         -= floorf(u);
                float pos  = u * (float)LUT_SIZE;
                int   i0   = (int)pos;
                float frac = pos - (float)i0;      // before mask: handles pos==4096
                i0        &= (LUT_SIZE - 1);
                float4 sc  = tab4[i0];             // single ds_load_b128 gather
                float s = fmaf(sc.z - sc.x, frac, sc.x);
                float c = fmaf(sc.w - sc.y, frac, sc.y);
                accc[bb] = fmaf(c, acw[j], accc[bb]);
                accs[bb] = fmaf(s, asw[j], accs[bb]);
            }
        }
    }

    // wave32 reductions
    #pragma unroll
    for (int off = 16; off > 0; off >>= 1) {
        #pragma unroll
        for (int bb = 0; bb < NB; ++bb) {
            accc[bb] += __shfl_down(accc[bb], off, 32);
            accs[bb] += __shfl_down(accs[bb], off, 32);
        }
    }
    if (lane == 0) {
        #pragma unroll
        for (int bb = 0; bb < NB; ++bb) {
            cos_sum[(size_t)(b0 + bb) * NUM_NEURONS + n] = accc[bb];
            sin_sum[(size_t)(b0 + bb) * NUM_NEURONS + n] = accs[bb];
        }
    }
}

// ---------------------------------------------------------------------------
// Kernel 3: out[b,j] = silu( sum[b,:] . w[j,:] )   (run twice: real & imag)
//   M=64, N=1024, K=2048 fp32 WMMA, same tiling as kernel 1.
// ---------------------------------------------------------------------------
__global__ void __launch_bounds__(128)
out_gemm_silu_kernel(const float* __restrict__ sum, const float* __restrict__ w,
                     float* __restrict__ out) {
    const int K = NUM_NEURONS;
    int wave = threadIdx.x >> 5;
    int lane = threadIdx.x & 31;
    int tile = blockIdx.x * 4 + wave;
    int tm = tile & 3;
    int tn = tile >> 2;
    int half = lane >> 4;
    int l16  = lane & 15;
    int arow = tm * 16 + l16;
    int bcol = tn * 16 + l16;

    v8f acc = {};
    for (int k0 = 0; k0 < K; k0 += 4) {
        int ka = k0 + half * 2;
        v2f a, b;
        const float* asrc = sum + (size_t)arow * K + ka;  // [64,2048] row-major
        a.x = asrc[0]; a.y = asrc[1];
        const float* bsrc = w + (size_t)bcol * K + ka;    // B[k,j] = w[j,k]
        b.x = bsrc[0]; b.y = bsrc[1];
        acc = __builtin_amdgcn_wmma_f32_16x16x4_f32(false, a, false, b,
                                                    (short)0, acc, false, false);
    }
    int mbase = tm * 16 + half * 8;
    #pragma unroll
    for (int r = 0; r < 8; ++r) {
        float v = acc[r];
        float sig = 1.0f / (1.0f + __expf(-v));
        out[(size_t)(mbase + r) * D_MODEL + bcol] = v * sig;
    }
}

// ---------------------------------------------------------------------------
extern "C" void kernel_launch(void* const* d_in, const int* in_sizes, int n_in,
                              void* d_out, int out_size, void* d_ws, size_t ws_size,
                              hipStream_t stream) {
    (void)in_sizes; (void)n_in; (void)out_size; (void)ws_size;
    const float* x_real   = (const float*)d_in[0];
    const float* x_imag   = (const float*)d_in[1];
    const float* t        = (const float*)d_in[2];
    const float* ic_w     = (const float*)d_in[3];
    const float* ic_b     = (const float*)d_in[4];
    const float* W        = (const float*)d_in[5];
    const float* Bp       = (const float*)d_in[6];
    const float* attn_cos = (const float*)d_in[7];
    const float* attn_sin = (const float*)d_in[8];
    const float* wr       = (const float*)d_in[9];
    const float* wi       = (const float*)d_in[10];
    const float* sin_tab  = (const float*)d_in[11];
    const float* cos_tab  = (const float*)d_in[12];

    float* ws = (float*)d_ws;
    float* xc = ws;                                    // 64*1024
    float* cs = ws + BATCH * D_MODEL;                  // 64*2048
    float* ss = cs + BATCH * NUM_NEURONS;              // 64*2048

    float* out_real = (float*)d_out;
    float* out_imag = out_real + BATCH * D_MODEL;

    // 256 tiles / 4 waves per block
    collapse_gemm_kernel<<<64, 128, 0, stream>>>(x_real, x_imag, ic_w, ic_b, xc);

    // (64/NB) batch groups x 256 neuron-groups
    resonant_core_kernel<<<(BATCH / NB) * 256, 256, 0, stream>>>(
        xc, t, W, Bp, attn_cos, attn_sin, sin_tab, cos_tab, cs, ss);

    out_gemm_silu_kernel<<<64, 128, 0, stream>>>(cs, wr, out_real);
    out_gemm_silu_kernel<<<64, 128, 0, stream>>>(ss, wi, out_imag);
}